// RNN_79620103733907
// MI455X (gfx1250) — compile-verified
//
#include <hip/hip_runtime.h>
#include <hip/hip_bf16.h>

// Problem dims (fixed by reference)
#define BATCH 2048
#define SEQ   512
#define DIN   128
#define HID   64
#define OUTD  8
#define BN_EPS 1e-5f

typedef __attribute__((ext_vector_type(2))) float v2f;
typedef __attribute__((ext_vector_type(8))) float v8f;

// V_WMMA_F32_16X16X4_F32 : D(16x16 f32) = A(16x4 f32) * B(4x16 f32) + C
// A,B = 2 VGPRs/lane ; C,D = 8 VGPRs/lane (wave32)
static __device__ __forceinline__ v8f wmma4(v2f a, v2f b, v8f c) {
  return __builtin_amdgcn_wmma_f32_16x16x4_f32(
      /*neg_a=*/false, a, /*neg_b=*/false, b,
      /*c_mod=*/(short)0, c, /*reuse_a=*/false, /*reuse_b=*/false);
}

// -------------------------------------------------------------------------
// Kernel 1: xp = x @ W_ih^T + (b_ih + b_hh), written in WMMA C-fragment
// layout: xp_frag[tile][ntile][lane][8], tile = (s*BATCH + b)/16.
// Grid: 8192 blocks x 256 thr (8 waves); each wave = one 16x64 output tile.
// -------------------------------------------------------------------------
__global__ __launch_bounds__(256) void k_input_proj(
    const float* __restrict__ x,      // [B,S,D]
    const float* __restrict__ W_ih,   // [H,D]
    const float* __restrict__ b_ih,
    const float* __restrict__ b_hh,
    float* __restrict__ xp_frag)      // [S*B/16][4][32][8] f32
{
  __shared__ float sW[HID * 132];     // padded stride 132 -> conflict-free
  __shared__ float sB[HID];

  const int tid = threadIdx.x;
  for (int i = tid; i < HID * DIN; i += 256) {
    int h = i >> 7, d = i & 127;
    sW[h * 132 + d] = W_ih[i];
  }
  if (tid < HID) sB[tid] = b_ih[tid] + b_hh[tid];
  __syncthreads();

  const int wave  = tid >> 5;
  const int lane  = tid & 31;
  const int lhalf = lane >> 4;        // K sub-pair selector
  const int l16   = lane & 15;

  const long tile = (long)blockIdx.x * 8 + wave;    // 65536 tiles of 16 rows
  const long r    = tile * 16 + l16;                // output row = s*BATCH + b
  const int  s    = (int)(r >> 11);                 // / 2048
  const int  b    = (int)(r & 2047);
  const float* xrow = x + ((long)b * SEQ + s) * DIN;

  v8f acc[4];
#pragma unroll
  for (int nt = 0; nt < 4; ++nt) {
    float bb = sB[l16 + nt * 16];
    acc[nt] = (v8f){bb, bb, bb, bb, bb, bb, bb, bb};
  }

#pragma unroll 8
  for (int kk = 0; kk < DIN / 4; ++kk) {            // K = 128, 4 per WMMA
    const int k = kk * 4 + lhalf * 2;
    v2f a = *(const v2f*)(xrow + k);                // A frag: M=l16, K={k,k+1}
#pragma unroll
    for (int nt = 0; nt < 4; ++nt) {
      const int n = l16 + nt * 16;
      v2f bf = *(const v2f*)(&sW[n * 132 + k]);     // B[k][n] = W_ih[n][k]
      acc[nt] = wmma4(a, bf, acc[nt]);
    }
  }

  float* outp = xp_frag + ((tile * 4) * 32 + lane) * 8;
#pragma unroll
  for (int nt = 0; nt < 4; ++nt)
    *(v8f*)(outp + nt * 32 * 8) = acc[nt];
}

// -------------------------------------------------------------------------
// Kernel 2: recurrent scan  h = relu(xp[t] + h @ W_hh^T), t = 0..511.
// 64 blocks x 64 thr (2 waves); each wave owns 16 batch rows.
// W_hh B-fragments live in 128 VGPRs; h round-trips through padded LDS
// (stride 68) for the C-layout -> A-layout swizzle each step.
// -------------------------------------------------------------------------
__global__ __launch_bounds__(64) void k_rnn_scan(
    const float* __restrict__ xp_frag,
    const float* __restrict__ W_hh,   // [H,H]
    const float* __restrict__ h0,     // [1,B,H]
    float* __restrict__ hT)           // [B,H]
{
  __shared__ float sh[2][16 * 68];

  const int tid   = threadIdx.x;
  const int wave  = tid >> 5;
  const int lane  = tid & 31;
  const int lhalf = lane >> 4;
  const int l16   = lane & 15;
  const int btile = blockIdx.x * 2 + wave;          // 0..127
  const int b0    = btile * 16;
  float* hld = sh[wave];

  // W_hh B-fragments: B[k][n] = W_hh[n][k]
  v2f wf[16][4];
#pragma unroll
  for (int kk = 0; kk < 16; ++kk)
#pragma unroll
    for (int nt = 0; nt < 4; ++nt) {
      const int n = l16 + nt * 16;
      const int k = kk * 4 + lhalf * 2;
      wf[kk][nt] = *(const v2f*)(W_hh + n * HID + k);
    }

  // h(0) from h0 (zeros in reference, but honor the input)
  for (int i = lane; i < 16 * HID; i += 32) {
    int m = i >> 6, n = i & 63;
    hld[m * 68 + n] = h0[(long)(b0 + m) * HID + n];
  }

  const long tile_stride = (long)(BATCH / 16) * 4 * 32 * 8;   // floats per t
  const float* xb = xp_frag + (long)btile * 4 * 32 * 8 + lane * 8;

  v8f cur[4], nxt[4];
#pragma unroll
  for (int nt = 0; nt < 4; ++nt)
    cur[nt] = *(const v8f*)(xb + nt * 256);

  for (int t = 0; t < SEQ; ++t) {
    // deep prefetch to cover HBM latency across the serial chain
    if (t + 6 < SEQ)
      __builtin_prefetch(xb + (long)(t + 6) * tile_stride, 0, 1);
    // software-pipelined next-step fragment load
    if (t + 1 < SEQ) {
      const float* xn = xb + (long)(t + 1) * tile_stride;
#pragma unroll
      for (int nt = 0; nt < 4; ++nt)
        nxt[nt] = *(const v8f*)(xn + nt * 256);
    }

    v8f a0 = cur[0], a1 = cur[1], a2 = cur[2], a3 = cur[3];
#pragma unroll
    for (int kk = 0; kk < 16; ++kk) {               // K = 64
      const int k = kk * 4 + lhalf * 2;
      v2f a = *(const v2f*)(&hld[l16 * 68 + k]);    // A frag of h
      a0 = wmma4(a, wf[kk][0], a0);
      a1 = wmma4(a, wf[kk][1], a1);
      a2 = wmma4(a, wf[kk][2], a2);
      a3 = wmma4(a, wf[kk][3], a3);
    }
#pragma unroll
    for (int i = 0; i < 8; ++i) {
      a0[i] = fmaxf(a0[i], 0.0f);
      a1[i] = fmaxf(a1[i], 0.0f);
      a2[i] = fmaxf(a2[i], 0.0f);
      a3[i] = fmaxf(a3[i], 0.0f);
    }

    // write new h back to LDS (C layout -> row major, conflict-free)
#pragma unroll
    for (int v = 0; v < 8; ++v) {
      const int m = v + lhalf * 8;
      hld[m * 68 + l16]      = a0[v];
      hld[m * 68 + l16 + 16] = a1[v];
      hld[m * 68 + l16 + 32] = a2[v];
      hld[m * 68 + l16 + 48] = a3[v];
    }

#pragma unroll
    for (int nt = 0; nt < 4; ++nt) cur[nt] = nxt[nt];
  }

  // final hidden state -> global, coalesced via LDS row-major copy
  for (int i = lane; i < 16 * HID; i += 32) {
    int m = i >> 6, n = i & 63;
    hT[(long)(b0 + m) * HID + n] = hld[m * 68 + n];
  }
}

// -------------------------------------------------------------------------
// Kernel 3a: BatchNorm1 batch statistics over hT, + zero bn2 accumulators.
// -------------------------------------------------------------------------
__global__ __launch_bounds__(256) void k_bn1_stats(
    const float* __restrict__ hT, float* __restrict__ stats1,
    float* __restrict__ sums2)
{
  __shared__ float ssum[256], ssq[256];
  const int tid = threadIdx.x;
  const int f = tid & 63, part = tid >> 6;
  float s = 0.f, q = 0.f;
  for (int r = part; r < BATCH; r += 4) {
    float v = hT[(long)r * HID + f];
    s += v; q += v * v;
  }
  ssum[tid] = s; ssq[tid] = q;
  __syncthreads();
  if (part == 0) {
    s = ssum[f] + ssum[f + 64] + ssum[f + 128] + ssum[f + 192];
    q = ssq[f] + ssq[f + 64] + ssq[f + 128] + ssq[f + 192];
    float mu  = s * (1.0f / BATCH);
    float var = q * (1.0f / BATCH) - mu * mu;
    stats1[2 * f]     = mu;
    stats1[2 * f + 1] = rsqrtf(var + BN_EPS);
  }
  if (tid < 32) sums2[tid] = 0.0f;   // bn2 sum/sumsq accumulators
}

// -------------------------------------------------------------------------
// Kernel 3b: z1 = relu(bn1(hT) @ fc1_W^T + fc1_b); accumulate bn2 stats.
// -------------------------------------------------------------------------
__global__ __launch_bounds__(256) void k_fc1(
    const float* __restrict__ hT, const float* __restrict__ stats1,
    const float* __restrict__ g1, const float* __restrict__ bb1,
    const float* __restrict__ fc1_W, const float* __restrict__ fc1_b,
    float* __restrict__ z1, float* __restrict__ sums2)
{
  __shared__ float sWf[16 * 64], sb[16], sscale[64], sshift[64], red[32];
  const int tid = threadIdx.x;
  for (int i = tid; i < 16 * 64; i += 256) sWf[i] = fc1_W[i];
  if (tid < 16) sb[tid] = fc1_b[tid];
  if (tid < 64) {
    float mu = stats1[2 * tid], rstd = stats1[2 * tid + 1];
    float sc = rstd * g1[tid];
    sscale[tid] = sc;
    sshift[tid] = bb1[tid] - mu * sc;
  }
  if (tid < 32) red[tid] = 0.0f;
  __syncthreads();

  const int row = blockIdx.x * 256 + tid;
  float hv[64];
#pragma unroll 8
  for (int j = 0; j < 64; ++j)
    hv[j] = hT[(long)row * HID + j] * sscale[j] + sshift[j];

#pragma unroll
  for (int o = 0; o < 16; ++o) {
    float acc = sb[o];
#pragma unroll 8
    for (int j = 0; j < 64; ++j) acc += sWf[o * 64 + j] * hv[j];
    acc = fmaxf(acc, 0.0f);
    z1[(long)row * 16 + o] = acc;
    atomicAdd(&red[o], acc);
    atomicAdd(&red[16 + o], acc * acc);
  }
  __syncthreads();
  if (tid < 32) atomicAdd(&sums2[tid], red[tid]);
}

// -------------------------------------------------------------------------
// Kernel 3c: finalize bn2 stats.
// -------------------------------------------------------------------------
__global__ void k_bn2_stats(const float* __restrict__ sums2,
                            float* __restrict__ stats2)
{
  const int f = threadIdx.x;
  if (f < 16) {
    float mu  = sums2[f] * (1.0f / BATCH);
    float var = sums2[16 + f] * (1.0f / BATCH) - mu * mu;
    stats2[2 * f]     = mu;
    stats2[2 * f + 1] = rsqrtf(var + BN_EPS);
  }
}

// -------------------------------------------------------------------------
// Kernel 3d: out = bn2(z1) @ fc2_W^T + fc2_b.
// -------------------------------------------------------------------------
__global__ __launch_bounds__(256) void k_fc2(
    const float* __restrict__ z1, const float* __restrict__ stats2,
    const float* __restrict__ g2, const float* __restrict__ bb2,
    const float* __restrict__ fc2_W, const float* __restrict__ fc2_b,
    float* __restrict__ out)
{
  __shared__ float sW2[8 * 16], sb2[8], sscale[16], sshift[16];
  const int tid = threadIdx.x;
  if (tid < 8 * 16) sW2[tid] = fc2_W[tid];
  if (tid < 8)  sb2[tid] = fc2_b[tid];
  if (tid < 16) {
    float mu = stats2[2 * tid], rstd = stats2[2 * tid + 1];
    float sc = rstd * g2[tid];
    sscale[tid] = sc;
    sshift[tid] = bb2[tid] - mu * sc;
  }
  __syncthreads();

  const int row = blockIdx.x * 256 + tid;
  float zv[16];
#pragma unroll
  for (int j = 0; j < 16; ++j)
    zv[j] = z1[(long)row * 16 + j] * sscale[j] + sshift[j];
#pragma unroll
  for (int o = 0; o < 8; ++o) {
    float acc = sb2[o];
#pragma unroll
    for (int j = 0; j < 16; ++j) acc += sW2[o * 16 + j] * zv[j];
    out[(long)row * OUTD + o] = acc;
  }
}

// -------------------------------------------------------------------------
extern "C" void kernel_launch(void* const* d_in, const int* in_sizes, int n_in,
                              void* d_out, int out_size, void* d_ws, size_t ws_size,
                              hipStream_t stream) {
  const float* x     = (const float*)d_in[0];
  const float* h0    = (const float*)d_in[1];
  const float* W_ih  = (const float*)d_in[2];
  const float* W_hh  = (const float*)d_in[3];
  const float* b_ih  = (const float*)d_in[4];
  const float* b_hh  = (const float*)d_in[5];
  const float* bn1_g = (const float*)d_in[6];
  const float* bn1_b = (const float*)d_in[7];
  const float* fc1_W = (const float*)d_in[8];
  const float* fc1_b = (const float*)d_in[9];
  const float* bn2_g = (const float*)d_in[10];
  const float* bn2_b = (const float*)d_in[11];
  const float* fc2_W = (const float*)d_in[12];
  const float* fc2_b = (const float*)d_in[13];

  float* out = (float*)d_out;                       // [B, O]
  float* hT  = (float*)d_out + (size_t)BATCH * OUTD; // [B, H] (== [1,B,H])

  // workspace layout (floats)
  float* ws      = (float*)d_ws;
  float* xp_frag = ws;                                   // S*B*H = 67,108,864
  float* z1      = xp_frag + (size_t)SEQ * BATCH * HID;  // B*16
  float* stats1  = z1 + (size_t)BATCH * 16;              // 128
  float* sums2   = stats1 + 128;                         // 32
  float* stats2  = sums2 + 32;                           // 32

  // 1) input projection GEMM (WMMA f32), fragment-layout output
  k_input_proj<<<(SEQ * BATCH) / (8 * 16), 256, 0, stream>>>(
      x, W_ih, b_ih, b_hh, xp_frag);

  // 2) serialized recurrent scan (WMMA f32, W_hh in registers)
  k_rnn_scan<<<(BATCH / 16) / 2, 64, 0, stream>>>(xp_frag, W_hh, h0, hT);

  // 3) head
  k_bn1_stats<<<1, 256, 0, stream>>>(hT, stats1, sums2);
  k_fc1<<<BATCH / 256, 256, 0, stream>>>(hT, stats1, bn1_g, bn1_b,
                                         fc1_W, fc1_b, z1, sums2);
  k_bn2_stats<<<1, 32, 0, stream>>>(sums2, stats2);
  k_fc2<<<BATCH / 256, 256, 0, stream>>>(z1, stats2, bn2_g, bn2_b,
                                         fc2_W, fc2_b, out);
}